// SparseMaxPool_11759620456729
// MI455X (gfx1250) — compile-verified
//
#include <hip/hip_runtime.h>
#include <cstdint>

typedef __attribute__((ext_vector_type(4))) float v4f;

static constexpr int N = 64;
static constexpr int ROWS_PER_BLOCK = 16;            // 256 threads, 16 lanes/row
// Allowed diagonal offsets: 0..15, 17..31 odd, 35..63 step 4
static constexpr unsigned long long OFF_MASK = 0x88888888AAAAFFFFULL;

__global__ __launch_bounds__(256) void sparse_maxpool_kernel(
    const float* __restrict__ x, float* __restrict__ out, int nrows) {
  __shared__ float srow[ROWS_PER_BLOCK * N];         // 4 KB
  const int tid = threadIdx.x;

  // ---- CDNA5 async tensor path: DMA 16 rows (1024 floats) of x into LDS ----
  {
    // Clamp so a (theoretical) partial last block never reads OOB.
    size_t gidx = (size_t)blockIdx.x * (ROWS_PER_BLOCK * N) + (size_t)tid * 4;
    const size_t last = (size_t)nrows * N - 4;
    if (gidx > last) gidx = last;
    unsigned lds_off = (unsigned)(uintptr_t)(&srow[0]) + (unsigned)tid * 16u;
    unsigned long long gaddr = (unsigned long long)(x + gidx);
    asm volatile("global_load_async_to_lds_b128 %0, %1, off"
                 :: "v"(lds_off), "v"(gaddr) : "memory");
    asm volatile("s_wait_asynccnt 0" ::: "memory");
  }
  __syncthreads();

  const int r  = tid >> 4;                           // row within block
  const int t  = tid & 15;                           // lane within row
  const int j0 = t * 4;                              // first owned column
  const size_t row = (size_t)blockIdx.x * ROWS_PER_BLOCK + r;
  if (row >= (size_t)nrows) return;

  const float* xr = &srow[r * N];
  float* orow = out + row * (size_t)(N * N);

  float m0, m1, m2, m3;
  m0 = m1 = m2 = m3 = -__builtin_inff();

#pragma unroll
  for (int i = N - 1; i >= 0; --i) {
    const float xi = xr[i];                          // LDS broadcast read
    if (j0 + 0 >= i) m0 = fmaxf(m0, xi);
    if (j0 + 1 >= i) m1 = fmaxf(m1, xi);
    if (j0 + 2 >= i) m2 = fmaxf(m2, xi);
    if (j0 + 3 >= i) m3 = fmaxf(m3, xi);
    // (OFF_MASK << i) is a literal per unrolled iteration; bit j set iff
    // j >= i and (j - i) is an allowed diagonal.
    const unsigned bits = (unsigned)((OFF_MASK << i) >> j0) & 0xFu;
    v4f v;
    v.x = (bits & 1u) ? m0 : 0.0f;
    v.y = (bits & 2u) ? m1 : 0.0f;
    v.z = (bits & 4u) ? m2 : 0.0f;
    v.w = (bits & 8u) ? m3 : 0.0f;
    // Write-once 256 MiB stream: non-temporal b128 store.
    __builtin_nontemporal_store(v, (v4f*)(orow + (size_t)i * N + j0));
  }
}

__global__ void sparse_maxpool_mask_kernel(float* __restrict__ m) {
  const int idx = blockIdx.x * blockDim.x + threadIdx.x;
  if (idx >= N * N) return;
  const int i = idx >> 6;
  const int j = idx & (N - 1);
  m[idx] = (j >= i && ((OFF_MASK >> (j - i)) & 1ull)) ? 1.0f : 0.0f;
}

extern "C" void kernel_launch(void* const* d_in, const int* in_sizes, int n_in,
                              void* d_out, int out_size, void* d_ws, size_t ws_size,
                              hipStream_t stream) {
  const float* x = (const float*)d_in[0];
  float* out = (float*)d_out;
  const int nrows = in_sizes[0] / N;                 // B * D = 16384
  const int blocks = (nrows + ROWS_PER_BLOCK - 1) / ROWS_PER_BLOCK;
  sparse_maxpool_kernel<<<blocks, 256, 0, stream>>>(x, out, nrows);
  float* maskp = out + (size_t)nrows * N * N;
  sparse_maxpool_mask_kernel<<<(N * N + 255) / 256, 256, 0, stream>>>(maskp);
}